// SparseLinear_83837761617924
// MI455X (gfx1250) — compile-verified
//
#include <hip/hip_runtime.h>
#include <stdint.h>

typedef __attribute__((ext_vector_type(2))) float v2f;
typedef __attribute__((ext_vector_type(4))) float v4f;
typedef __attribute__((ext_vector_type(8))) float v8f;
typedef __attribute__((ext_vector_type(4))) unsigned int v4u;
typedef __attribute__((ext_vector_type(8))) int v8i;
typedef __attribute__((ext_vector_type(4))) int v4i;

#define IN_DIMS 8192
#define OUT_DIMS 8192
#define KC 128              // K-chunk staged in LDS per iteration
#define WSTRIDE (KC + 4)    // padded LDS row stride (floats): conflict-free b64 reads
#define NCHUNK (IN_DIMS / KC)

#if __has_builtin(__builtin_amdgcn_tensor_load_to_lds) && \
    __has_builtin(__builtin_amdgcn_s_wait_tensorcnt)
#define HAVE_TDM 1
#else
#define HAVE_TDM 0
#endif

#if HAVE_TDM
// One TDM descriptor loads a 16x128-float tile of W (row stride 8192 floats)
// into LDS, inserting 4 pad DWORDs after every 128 DWORDs -> LDS row stride 132.
__device__ __forceinline__ void tdm_load_tile(uint32_t lds_addr, uint64_t gaddr) {
  // group0: count=1 | lds_addr | global_addr[56:0] | type=2 (bits 127:126)
  v4u g0 = { 1u, lds_addr, (uint32_t)gaddr,
             ((uint32_t)(gaddr >> 32) & 0x01FFFFFFu) | (2u << 30) };
  // group1: data_size=4B(2<<16) | pad_enable(1<<20) | pad_interval=128DW(6<<22) |
  //         pad_amount=4DW(3<<25); tensor_dim0=8192; tensor_dim1=8192;
  //         tile_dim0=128; tile_dim1=16; tensor_dim0_stride=8192
  v8i g1 = { (int)((2u << 16) | (1u << 20) | (6u << 22) | (3u << 25)),
             (int)(((unsigned)IN_DIMS & 0xFFFFu) << 16),
             (int)(((unsigned)IN_DIMS >> 16) | (((unsigned)OUT_DIMS & 0xFFFFu) << 16)),
             (int)(((unsigned)OUT_DIMS >> 16) | ((unsigned)KC << 16)),
             16,            // tile_dim1=16, tile_dim2=0
             IN_DIMS,       // tensor_dim0_stride low 32
             0, 0 };
  v4i gz = { 0, 0, 0, 0 };  // 2D tensor: groups 2/3 unused
#if defined(__clang_major__) && __clang_major__ >= 23
  v8i gz8 = { 0, 0, 0, 0, 0, 0, 0, 0 };
  __builtin_amdgcn_tensor_load_to_lds(g0, g1, gz, gz, gz8, 0);
#else
  __builtin_amdgcn_tensor_load_to_lds(g0, g1, gz, gz, 0);
#endif
}
#endif

// out = W @ x + bias. One block (4 waves) -> 16 outputs. Per chunk: TDM stages
// the 16xKC W tile into LDS (double-buffered, overlapped with compute), then
// each wave runs 8x V_WMMA_F32_16X16X4_F32 with A = x broadcast, B = W^T tile.
__global__ __launch_bounds__(128) void sparse_linear_matvec_wmma(
    const float* __restrict__ W, const float* __restrict__ bias,
    const float* __restrict__ x, float* __restrict__ out) {

  __shared__ float lW[2][16 * WSTRIDE];
  __shared__ float lX[2][KC];
  __shared__ float lRed[64];

  const int tid  = threadIdx.x;
  const int wave = tid >> 5;
  const int lane = tid & 31;
  const int nbase = blockIdx.x << 4;

  // WMMA operand addressing (wave32): lanes 0-15 -> K {0,1}, lanes 16-31 -> K {2,3}
  const int row = lane & 15;         // N index for B-matrix
  const int hs  = (lane >> 4) << 1;  // 0 or 2

  v8f acc = {};  // C/D accumulator (16x16 f32 across the wave)

#if HAVE_TDM
  const uint64_t gW0   = (uint64_t)(uintptr_t)(W + (size_t)nbase * IN_DIMS);
  const uint32_t ldsW0 = (uint32_t)(uintptr_t)&lW[0][0];
  const uint32_t ldsW1 = (uint32_t)(uintptr_t)&lW[1][0];

  // Prologue: TDM chunk 0 -> buf0 (single issue per block: wave 0, per-wave DMA)
  if (wave == 0) tdm_load_tile(ldsW0, gW0);
  if (tid < (KC >> 2))
    *(v4f*)(&lX[0][tid << 2]) = *(const v4f*)(x + (tid << 2));

  for (int i = 0; i < NCHUNK; ++i) {
    const int cur = i & 1;
    if (i + 1 < NCHUNK) {
      // Overlap: issue next chunk's DMA into the other buffer, then wait for
      // the *older* DMA only (in-order per-wave completion => cnt<=1 is chunk i).
      if (wave == 0) tdm_load_tile(cur ? ldsW0 : ldsW1,
                                   gW0 + (uint64_t)(i + 1) * KC * sizeof(float));
      if (tid < (KC >> 2))
        *(v4f*)(&lX[1 - cur][tid << 2]) =
            *(const v4f*)(x + (i + 1) * KC + (tid << 2));
      if (i + 2 < NCHUNK)  // warm GL2 for the chunk after next
        __builtin_prefetch(W + (size_t)(nbase + (tid >> 3)) * IN_DIMS +
                               (i + 2) * KC + ((tid & 7) << 4), 0, 0);
      if (wave == 0) __builtin_amdgcn_s_wait_tensorcnt((short)1);
    } else {
      if (wave == 0) __builtin_amdgcn_s_wait_tensorcnt((short)0);
    }
    __syncthreads();

    const int kw = wave << 5;
#pragma unroll
    for (int kk = 0; kk < 32; kk += 4) {
      v2f a = *(const v2f*)(&lX[cur][kw + kk + hs]);                  // A: x bcast
      v2f b = *(const v2f*)(&lW[cur][row * WSTRIDE + kw + kk + hs]);  // B: W^T tile
      acc = __builtin_amdgcn_wmma_f32_16x16x4_f32(false, a, false, b,
                                                  (short)0, acc, false, false);
    }
    __syncthreads();
  }
#else
  // Fallback: manual coalesced staging (round-1 path, single buffer)
  const int sr = tid >> 3;
  const int sc = tid & 7;
  const float* gW = W + (size_t)(nbase + sr) * IN_DIMS;
  for (int k0 = 0; k0 < IN_DIMS; k0 += KC) {
#pragma unroll
    for (int j = 0; j < 4; ++j) {
      const int c4 = (sc + (j << 3)) << 2;
      *(v4f*)(&lW[0][sr * WSTRIDE + c4]) = *(const v4f*)(gW + k0 + c4);
    }
    if (tid < (KC >> 2))
      *(v4f*)(&lX[0][tid << 2]) = *(const v4f*)(x + k0 + (tid << 2));
    if (k0 + KC < IN_DIMS)
      __builtin_prefetch(gW + k0 + KC + (sc << 4), 0, 0);
    __syncthreads();
    const int kw = wave << 5;
#pragma unroll
    for (int kk = 0; kk < 32; kk += 4) {
      v2f a = *(const v2f*)(&lX[0][kw + kk + hs]);
      v2f b = *(const v2f*)(&lW[0][row * WSTRIDE + kw + kk + hs]);
      acc = __builtin_amdgcn_wmma_f32_16x16x4_f32(false, a, false, b,
                                                  (short)0, acc, false, false);
    }
    __syncthreads();
  }
#endif

  // D row M=0 lives in acc[0] on lanes 0-15 (N = lane). Reduce the 4 waves'
  // K-partition partials through LDS, add bias, store.
  if (lane < 16) lRed[(wave << 4) + lane] = acc[0];
  __syncthreads();
  if (tid < 16) {
    float s = lRed[tid] + lRed[16 + tid] + lRed[32 + tid] + lRed[48 + tid];
    out[nbase + tid] = s + bias[nbase + tid];
  }
}

extern "C" void kernel_launch(void* const* d_in, const int* in_sizes, int n_in,
                              void* d_out, int out_size, void* d_ws, size_t ws_size,
                              hipStream_t stream) {
  (void)in_sizes; (void)n_in; (void)d_ws; (void)ws_size; (void)out_size;
  const float* W    = (const float*)d_in[0];
  const float* bias = (const float*)d_in[1];
  const float* x    = (const float*)d_in[2];
  float* out        = (float*)d_out;

  dim3 grid(OUT_DIMS / 16);  // 512 blocks
  dim3 block(128);           // 4 waves (wave32)
  hipLaunchKernelGGL(sparse_linear_matvec_wmma, grid, block, 0, stream,
                     W, bias, x, out);
}